// MemoryBankCompressor_24446953849506
// MI455X (gfx1250) — compile-verified
//
#include <hip/hip_runtime.h>
#include <hip/hip_bf16.h>

typedef __attribute__((ext_vector_type(2))) float v2f;
typedef __attribute__((ext_vector_type(4))) float v4f;
typedef __attribute__((ext_vector_type(8))) float v8f;

namespace {
constexpr int kB = 4;
constexpr int kL = 64;
constexpr int kN = 256;
constexpr int kC = 1024;
constexpr int kLm1 = kL - 1;
constexpr int kTiles = 5;               // tile starts cover all 63 adjacent pairs
constexpr int kSlice4 = kN * kC / 4;    // float4 per (b,l) slice = 65536
}

// ---------------------------------------------------------------------------
// Kernel 1: per (b, n, tileStart) wave32 computes the 16x16 Gram matrix of 16
// consecutive L-rows over C=1024 using V_WMMA_F32_16X16X4_F32 (256 MACC steps).
// A-matrix 16x4 layout: lane = kh*16 + m holds row m, K = 4*s + 2*kh + {0,1}.
// B-matrix 4x16 layout is the same striping, so the same float2 register is
// fed as both A and B -> D[i][j] += sum_k row_i[k]*row_j[k]  (Gram tile).
// Diagonal -> norm^2, superdiagonal -> dot(l, l+1); emit cosine sims directly.
// ---------------------------------------------------------------------------
__global__ __launch_bounds__(32) void gram_sims_kernel(
    const float* __restrict__ bank, float* __restrict__ sims) {
  const int lane = threadIdx.x;
  int id = blockIdx.x;
  const int t = id % kTiles; id /= kTiles;
  const int n = id % kN;     id /= kN;
  const int b = id;

  const int tsArr[kTiles] = {0, 15, 30, 45, 48};
  const int ts = tsArr[t];

  const int m  = lane & 15;   // row within tile
  const int kh = lane >> 4;   // K-half selector
  const float* rowp = bank
      + (((size_t)(b * kL + ts + m) * kN + n) * kC) + 2 * kh;

  v8f acc = {0.f, 0.f, 0.f, 0.f, 0.f, 0.f, 0.f, 0.f};
#pragma unroll 8
  for (int s = 0; s < kC / 4; ++s) {
    v2f a = *(const v2f*)(rowp + 4 * s);   // global_load_b64, 8B aligned
    acc = __builtin_amdgcn_wmma_f32_16x16x4_f32(
        /*neg_a=*/false, a, /*neg_b=*/false, a,
        /*c_mod=*/(short)0, acc, /*reuse_a=*/false, /*reuse_b=*/false);
  }

  // C/D layout: VGPR r, lanes 0-15 -> (M=r, N=lane); lanes 16-31 -> (M=8+r).
  __shared__ float G[16][17];
  const int Mb = kh * 8;
#pragma unroll
  for (int r = 0; r < 8; ++r) G[Mb + r][m] = acc[r];
  __syncthreads();

  if (lane < 15) {
    const int i = lane;                    // pair (ts+i, ts+i+1)
    const float d  = G[i][i + 1];
    const float na = fmaxf(sqrtf(G[i][i]),         1e-8f);
    const float nb = fmaxf(sqrtf(G[i + 1][i + 1]), 1e-8f);
    sims[((size_t)b * kLm1 + (ts + i)) * kN + n] = d / (na * nb);
  }
}

// ---------------------------------------------------------------------------
// Kernel 2: per batch, mean over N and first-occurrence argmax over l.
// ---------------------------------------------------------------------------
__global__ __launch_bounds__(256) void argmax_kernel(
    const float* __restrict__ sims, int* __restrict__ kOut) {
  const int b = blockIdx.x;
  const int t = threadIdx.x;
  __shared__ float red[256];

  float best = -3.402823e38f;
  int bestl = 0;
  for (int l = 0; l < kLm1; ++l) {
    red[t] = sims[((size_t)b * kLm1 + l) * kN + t];
    __syncthreads();
#pragma unroll
    for (int off = 128; off > 0; off >>= 1) {
      if (t < off) red[t] += red[t + off];
      __syncthreads();
    }
    if (t == 0 && red[0] > best) { best = red[0]; bestl = l; }
    __syncthreads();
  }
  if (t == 0) kOut[b] = bestl;
}

// ---------------------------------------------------------------------------
// Kernel 3: streaming gather/merge, float4 vectorized, non-temporal (no reuse).
// out[b,i] = (i==k) ? 0.5*(bank[b,k]+bank[b,k+1]) : bank[b, i + (i>=k)]
// ---------------------------------------------------------------------------
__global__ __launch_bounds__(256) void gather_merge_kernel(
    const float* __restrict__ bank, const int* __restrict__ kArr,
    float* __restrict__ out) {
  const size_t idx = (size_t)blockIdx.x * blockDim.x + threadIdx.x;
  const size_t per_b = (size_t)kLm1 * kSlice4;
  const int b = (int)(idx / per_b);
  const size_t rem = idx - (size_t)b * per_b;
  const int i   = (int)(rem / kSlice4);
  const int off = (int)(rem - (size_t)i * kSlice4);
  const int k = kArr[b];

  const v4f* bank4 = (const v4f*)bank;
  v4f v;
  if (i == k) {
    v4f x = __builtin_nontemporal_load(
        &bank4[(size_t)(b * kL + k) * kSlice4 + off]);
    v4f y = __builtin_nontemporal_load(
        &bank4[(size_t)(b * kL + k + 1) * kSlice4 + off]);
    v = (x + y) * 0.5f;
  } else {
    const int src = i + (i >= k ? 1 : 0);
    v = __builtin_nontemporal_load(
        &bank4[(size_t)(b * kL + src) * kSlice4 + off]);
  }
  __builtin_nontemporal_store(v, (v4f*)out + idx);
}

// ---------------------------------------------------------------------------
extern "C" void kernel_launch(void* const* d_in, const int* in_sizes, int n_in,
                              void* d_out, int out_size, void* d_ws, size_t ws_size,
                              hipStream_t stream) {
  const float* bank = (const float*)d_in[0];
  float* out = (float*)d_out;

  // Workspace layout: sims[B][63][N] floats, then k[B] ints.
  float* sims = (float*)d_ws;
  int*   kArr = (int*)((char*)d_ws + (size_t)kB * kLm1 * kN * sizeof(float));

  // Pass 1: Gram-matrix cosine sims via f32 WMMA. One wave per (b, n, tile).
  gram_sims_kernel<<<kB * kN * kTiles, 32, 0, stream>>>(bank, sims);

  // Pass 2: mean over N + argmax over l, one block per batch.
  argmax_kernel<<<kB, 256, 0, stream>>>(sims, kArr);

  // Pass 3: streaming gather/merge. Grid divides exactly: 4*63*65536/256.
  const int total4 = kB * kLm1 * kSlice4;
  gather_merge_kernel<<<total4 / 256, 256, 0, stream>>>(bank, kArr, out);
}